// PSP_29265907155564
// MI455X (gfx1250) — compile-verified
//
#include <hip/hip_runtime.h>
#include <math.h>
#include <stdint.h>

typedef __attribute__((ext_vector_type(2))) float v2f;
typedef __attribute__((ext_vector_type(4))) float v4f;
typedef __attribute__((ext_vector_type(8))) float v8f;

#define T_DIM 128
#define ROW_PAD 132          // 128 + 4 floats: conflict-free A-fragment ds_load_b64
#define WAVES_PER_BLOCK 4
#define ROWS_PER_WAVE 16
#define ROWS_PER_BLOCK (WAVES_PER_BLOCK * ROWS_PER_WAVE)
#define USE_ASYNC_LDS 1      // gfx1250 async global->LDS staging (VGPR bypass)

// out[r, t] = sum_{k<=t} 0.5^(t-k+1) * x[r, k]   (closed form of the leaky scan)
// Banded to k >= 16*floor(t/16) - 32 (dropped weights <= 2^-34: below fp32 epsilon).
__global__ __launch_bounds__(WAVES_PER_BLOCK * 32)
void psp_wmma_kernel(const float* __restrict__ x, float* __restrict__ out, int rows) {
    __shared__ float lds[WAVES_PER_BLOCK * ROWS_PER_WAVE * ROW_PAD];

    const int lane  = threadIdx.x & 31;
    const int wave  = threadIdx.x >> 5;
    const int row_base = blockIdx.x * ROWS_PER_BLOCK + wave * ROWS_PER_WAVE;
    const bool active = (row_base + ROWS_PER_WAVE) <= rows;

    float* wlds = lds + wave * ROWS_PER_WAVE * ROW_PAD;

    if (!active) return;  // wave-private staging: no cross-wave barrier needed

    const int mrow  = lane & 15;   // A-row (M); also B/D column (N)
    const int khalf = lane >> 4;   // 0: K={0,1}; 1: K={2,3} within a 16x4 tile

    // ---- Stage 16 rows x 128 floats into LDS (coalesced 512B per row) ----
    const float* src = x + (size_t)row_base * T_DIM;
#if USE_ASYNC_LDS
    #pragma unroll
    for (int m = 0; m < ROWS_PER_WAVE; ++m) {
        unsigned long long gaddr =
            (unsigned long long)(uintptr_t)(src + m * T_DIM + lane * 4);
        unsigned int laddr =
            (unsigned int)(uintptr_t)(wlds + m * ROW_PAD + lane * 4);  // flat low32 == LDS byte addr
        asm volatile("global_load_async_to_lds_b128 %0, %1, off"
                     :: "v"(laddr), "v"(gaddr) : "memory");
    }
    asm volatile("s_wait_asynccnt 0x0" ::: "memory");
#else
    #pragma unroll
    for (int m = 0; m < ROWS_PER_WAVE; ++m) {
        v4f v = *(const v4f*)(src + m * T_DIM + lane * 4);
        *(v4f*)(wlds + m * ROW_PAD + lane * 4) = v;
    }
#endif

    // ---- Precompute the only 3 distinct B supertiles (d = j - i = 0,1,2) ----
    // B[k_local, n] with value 2^(k-t-1), k-t = koff - 16d - mrow; zero if k > t.
    v2f bfrag[3][4];
    #pragma unroll
    for (int d = 0; d < 3; ++d) {
        #pragma unroll
        for (int q = 0; q < 4; ++q) {
            const int koff = q * 4 + khalf * 2;
            const int dlt  = koff - 16 * d - mrow;  // k - t for element .x
            v2f b;
            b.x = (dlt     <= 0) ? ldexpf(0.5f, dlt)     : 0.0f;
            b.y = (dlt + 1 <= 0) ? ldexpf(0.5f, dlt + 1) : 0.0f;
            bfrag[d][q] = b;
        }
    }

    // ---- Main loop: rotating 3-deep A-supertile register buffer ----
    v2f afrag[3][4];
    float* orow = out + (size_t)row_base * T_DIM;
    const float* arow = wlds + mrow * ROW_PAD + khalf * 2;

    #pragma unroll
    for (int j = 0; j < 8; ++j) {           // output supertile: t in [16j, 16j+15]
        // load A supertile j (each supertile loaded exactly once)
        #pragma unroll
        for (int q = 0; q < 4; ++q)
            afrag[j % 3][q] = *(const v2f*)(arow + j * 16 + q * 4);

        v8f acc = {};
        const int i0 = (j >= 2) ? (j - 2) : 0;  // causal band: 48 taps
        #pragma unroll
        for (int i = i0; i <= j; ++i) {
            const int d = j - i;
            #pragma unroll
            for (int q = 0; q < 4; ++q) {
                acc = __builtin_amdgcn_wmma_f32_16x16x4_f32(
                    false, afrag[i % 3][q], false, bfrag[d][q],
                    (short)0, acc, false, false);
            }
        }

        // D layout: VGPR v, lanes 0-15 -> M=v, lanes 16-31 -> M=v+8; N = lane&15
        #pragma unroll
        for (int v = 0; v < 8; ++v) {
            const int m = v + khalf * 8;
            orow[(size_t)m * T_DIM + j * 16 + mrow] = acc[v];
        }
    }
}

extern "C" void kernel_launch(void* const* d_in, const int* in_sizes, int n_in,
                              void* d_out, int out_size, void* d_ws, size_t ws_size,
                              hipStream_t stream) {
    const float* x = (const float*)d_in[0];
    float* out = (float*)d_out;

    const int total = in_sizes[0];          // N*C*T = 256*1024*128
    const int rows  = total / T_DIM;        // 262144 rows of length 128
    const int grid  = (rows + ROWS_PER_BLOCK - 1) / ROWS_PER_BLOCK;  // 4096

    psp_wmma_kernel<<<grid, WAVES_PER_BLOCK * 32, 0, stream>>>(x, out, rows);
}